// LightweightConv1dTBC_54881092108718
// MI455X (gfx1250) — compile-verified
//
#include <hip/hip_runtime.h>

// LightweightConv1dTBC on MI455X (gfx1250, wave32).
// Depthwise 31-tap conv as banded-Toeplitz matmul on the matrix pipe:
//   V_WMMA_F32_16X16X4_F32, full fp32 precision.
// Memory-bound: 128 MB HBM traffic @ 23.3 TB/s ~= 5.5 us floor; x (64 MB)
// fits in the 192 MB L2 so halo re-reads are free. Setup kernel precomputes
// softmax taps AND the per-(head,lane) Toeplitz A-fragments so the hot kernel
// is just: 12 coalesced b64 A-loads -> 4 tiles x (24 b32 B-loads + 12 WMMA).

typedef __attribute__((ext_vector_type(2))) float v2f;
typedef __attribute__((ext_vector_type(8))) float v8f;

#define T_DIM    2048
#define B_DIM    8
#define C_DIM    1024
#define N_DIM    (B_DIM * C_DIM)     // 8192 flattened (b,c) columns
#define H_DIM    16
#define K_TAPS   31
#define PAD_L    15
#define NCHUNK   12                  // K band 46 -> padded 48 -> 12 chunks of 4

#define TILES_N  (N_DIM / 16)        // 512
#define TILES_T  (T_DIM / 16)        // 128
#define TT_PER_WAVE 4                // T-tiles per wave (share A fragments)
#define TILES_G  (TILES_T / TT_PER_WAVE)  // 32 t-groups
#define WAVES_PER_BLOCK 8

// ---------------------------------------------------------------------------
// Setup kernel: softmax(weight) per head (in LDS), then materialize the
// Toeplitz A-fragment table a_tab[h][j][lane] : v2f   (16*12*32*8B = 48 KB).
//   A[m, k] = sw_h[k - m] for 0 <= k-m <= 30 else 0, per WMMA f32 16x4 layout:
//   lane: m = lane&15, K-pair = (lane>>4)*2 + {0,1} within chunk j (K=4j..4j+3)
// ---------------------------------------------------------------------------
__global__ __launch_bounds__(512)
void lconv_setup_kernel(const float* __restrict__ w, float* __restrict__ ws) {
    __shared__ float s_sw[H_DIM * 32];
    const int tid = threadIdx.x;

    if (tid < H_DIM) {
        const int h = tid;
        float m = -3.4e38f;
        #pragma unroll
        for (int k = 0; k < K_TAPS; ++k) m = fmaxf(m, w[h * K_TAPS + k]);
        float e[K_TAPS];
        float s = 0.0f;
        #pragma unroll
        for (int k = 0; k < K_TAPS; ++k) {
            e[k] = __expf(w[h * K_TAPS + k] - m);
            s += e[k];
        }
        const float inv = 1.0f / s;
        #pragma unroll
        for (int k = 0; k < K_TAPS; ++k) s_sw[h * 32 + k] = e[k] * inv;
        s_sw[h * 32 + K_TAPS] = 0.0f;
    }
    __syncthreads();

    const int h    = tid >> 5;         // head 0..15
    const int lane = tid & 31;
    const int lo   = lane & 15;        // A row m
    const int kbase = (lane >> 4) * 2; // K pair within chunk

    v2f* a_tab = (v2f*)ws;
    #pragma unroll
    for (int j = 0; j < NCHUNK; ++j) {
        const int i0 = 4 * j + kbase;
        const int d0 = i0 - lo;
        const int d1 = d0 + 1;
        v2f a;
        a.x = ((unsigned)d0 < (unsigned)K_TAPS) ? s_sw[h * 32 + d0] : 0.0f;
        a.y = ((unsigned)d1 < (unsigned)K_TAPS) ? s_sw[h * 32 + d1] : 0.0f;
        a_tab[(h * NCHUNK + j) * 32 + lane] = a;
    }
}

// ---------------------------------------------------------------------------
// Main kernel: one wave computes 4 consecutive 16(T) x 16(N) output tiles.
// ---------------------------------------------------------------------------
__global__ __launch_bounds__(256)
void lconv_wmma_kernel(const float* __restrict__ x,
                       const float* __restrict__ ws,
                       const float* __restrict__ bias,
                       float* __restrict__ out) {
    const v2f* __restrict__ a_tab = (const v2f*)ws;

    const int lane = threadIdx.x & 31;
    const int wave = threadIdx.x >> 5;
    const int id   = blockIdx.x * WAVES_PER_BLOCK + wave;

    const int tn  = id % TILES_N;          // n-tile 0..511
    const int tg  = id / TILES_N;          // t-group 0..31
    const int n0  = tn * 16;
    const int t0g = tg * (16 * TT_PER_WAVE);

    const int half  = lane >> 4;
    const int lo    = lane & 15;
    const int kbase = half * 2;

    const int col = n0 + lo;               // flattened (b,c) column
    const int c   = col & (C_DIM - 1);
    const int h   = c >> 6;                // head (uniform across tile)

    // A fragments: 12 coalesced, unpredicated b64 loads (reused for 4 tiles).
    v2f a[NCHUNK];
    #pragma unroll
    for (int j = 0; j < NCHUNK; ++j)
        a[j] = a_tab[(h * NCHUNK + j) * 32 + lane];

    const float* __restrict__ xcol = x + col;
    float* __restrict__ ocol = out + col;
    const float bv = bias[c];

    if (tg != 0 && tg != TILES_G - 1) {
        // ---- interior fast path: no boundary possible, raw loads ----------
        #pragma unroll
        for (int s = 0; s < TT_PER_WAVE; ++s) {
            const int t0 = t0g + s * 16;
            v8f acc = {};
            #pragma unroll
            for (int j = 0; j < NCHUNK; ++j) {
                const int tr0 = t0 - PAD_L + 4 * j + kbase;
                v2f b;
                b.x = xcol[tr0 * N_DIM];
                b.y = xcol[(tr0 + 1) * N_DIM];
                acc = __builtin_amdgcn_wmma_f32_16x16x4_f32(
                    false, a[j], false, b, (short)0, acc, false, false);
            }
            #pragma unroll
            for (int v = 0; v < 8; ++v)
                __builtin_nontemporal_store(acc[v] + bv,
                    &ocol[(t0 + v + half * 8) * N_DIM]);
        }
    } else {
        // ---- boundary path: clamp row (always-legal address) + select-0 ---
        #pragma unroll
        for (int s = 0; s < TT_PER_WAVE; ++s) {
            const int t0 = t0g + s * 16;
            v8f acc = {};
            #pragma unroll
            for (int j = 0; j < NCHUNK; ++j) {
                const int tr0 = t0 - PAD_L + 4 * j + kbase;
                const int tr1 = tr0 + 1;
                const int tc0 = min(max(tr0, 0), T_DIM - 1);
                const int tc1 = min(max(tr1, 0), T_DIM - 1);
                float v0 = xcol[tc0 * N_DIM];   // unconditional load
                float v1 = xcol[tc1 * N_DIM];
                v2f b;
                b.x = ((unsigned)tr0 < (unsigned)T_DIM) ? v0 : 0.0f;  // cndmask
                b.y = ((unsigned)tr1 < (unsigned)T_DIM) ? v1 : 0.0f;
                acc = __builtin_amdgcn_wmma_f32_16x16x4_f32(
                    false, a[j], false, b, (short)0, acc, false, false);
            }
            #pragma unroll
            for (int v = 0; v < 8; ++v)
                __builtin_nontemporal_store(acc[v] + bv,
                    &ocol[(t0 + v + half * 8) * N_DIM]);
        }
    }
}

// ---------------------------------------------------------------------------
extern "C" void kernel_launch(void* const* d_in, const int* in_sizes, int n_in,
                              void* d_out, int out_size, void* d_ws, size_t ws_size,
                              hipStream_t stream) {
    const float* x    = (const float*)d_in[0];  // (T, B, C) fp32
    const float* w    = (const float*)d_in[1];  // (H, 1, K) fp32
    const float* bias = (const float*)d_in[2];  // (C,) fp32
    float* out = (float*)d_out;                 // (T, B, C) fp32
    float* ws  = (float*)d_ws;                  // A-fragment table, 48 KB

    lconv_setup_kernel<<<1, 512, 0, stream>>>(w, ws);

    const int total_waves = TILES_N * TILES_G;            // 512 * 32 = 16384
    const int blocks = total_waves / WAVES_PER_BLOCK;     // 2048
    lconv_wmma_kernel<<<blocks, 256, 0, stream>>>(x, ws, bias, out);
}